// WindowEmbeddingforword_7086696038877
// MI455X (gfx1250) — compile-verified
//
#include <hip/hip_runtime.h>

// WindowEmbedding forward: out[b, t, k*D + d] = in[b, t-k, d] (0 if t-k < 0)
// B=16, T=2048, D=256, W=7, fp32.
// Pure data movement: ~33.5 MB read + ~235 MB write => write-BW bound,
// ~10 us floor at 23.3 TB/s. Stage a (TTILE + W-1)-row input tile in LDS via
// gfx1250 async global->LDS b128 DMA (ASYNCcnt), then emit the 7 shifted
// copies as straight-line, immediate-offset, fully coalesced b128 stores.

#define BB 16
#define TT 2048
#define DD 256
#define WW 7
#define TTILE 16
#define LROWS (TTILE + WW - 1)        // 22 staged rows
#define D4 (DD / 4)                   // 64 float4 per row
#define LDS_SLOTS (LROWS * D4)        // 1408 float4 = 22.5 KB
#define NTHREADS 256

// The async-LDS builtin takes v4i (int ext_vector(4)) pointers: AS1 src, AS3 dst.
typedef int v4i __attribute__((ext_vector_type(4)));
typedef __attribute__((address_space(1))) v4i gv4i;
typedef __attribute__((address_space(3))) v4i lv4i;

__global__ __launch_bounds__(NTHREADS)
void window_embed_fwd_kernel(const float* __restrict__ in, float* __restrict__ out) {
    __shared__ float4 tile[LDS_SLOTS];

    const int tid = threadIdx.x;
    const int d4  = tid & (D4 - 1);     // fixed column (float4) per thread: 0..63
    const int rg  = tid >> 6;           // row group: 0..3
    const int blk = blockIdx.x;
    const int b   = blk >> 7;                       // / (TT/TTILE) = /128
    const int tt  = blk & ((TT / TTILE) - 1);       // % 128
    const int t0  = tt * TTILE - (WW - 1);          // first staged timestep (may be < 0)

    const float4* __restrict__ in4 = reinterpret_cast<const float4*>(in);
    // Global source for this thread's column at staged row 0:
    const float4* gbase = &in4[((size_t)(b * TT + t0)) * D4 + d4];

    // ---- Phase 1: stage rows [t0, t0+22) into LDS (async b128 DMA), zero t<0 ----
    // Thread covers rows rg, rg+4, ..., rg+20 (rows 22/23 masked off).
    #pragma unroll
    for (int j = 0; j < 6; ++j) {
        const int row = rg + 4 * j;
        if (row < LROWS) {
            const int t = t0 + row;
            if (t >= 0) {
                __builtin_amdgcn_global_load_async_to_lds_b128(
                    (gv4i*)(gbase + (size_t)row * D4),
                    (lv4i*)&tile[row * D4 + d4],
                    /*offset=*/0, /*cpol=*/0);
            } else {
                tile[row * D4 + d4] = make_float4(0.f, 0.f, 0.f, 0.f);
            }
        }
    }

    // Each wave waits for its own async DMAs, then cross-wave barrier.
#if __has_builtin(__builtin_amdgcn_s_wait_asynccnt)
    __builtin_amdgcn_s_wait_asynccnt(0);
#else
    asm volatile("s_wait_asynccnt 0" ::: "memory");
#endif
    __syncthreads();

    // ---- Phase 2: 28 straight-line b128 stores per thread, immediate offsets ----
    // out[b, tbase+tl, k*D + 4*d4] = tile[(tl + 6 - k) * D4 + d4]
    float4* __restrict__ obase =
        reinterpret_cast<float4*>(out) +
        ((size_t)(b * TT + tt * TTILE)) * (WW * D4) + d4;

    #pragma unroll
    for (int j = 0; j < 4; ++j) {
        const int tl = rg + 4 * j;                 // 0..15 across row groups
        #pragma unroll
        for (int k = 0; k < WW; ++k) {
            const float4 v = tile[(tl + (WW - 1) - k) * D4 + d4];  // imm-offset ds_load
            obase[(size_t)(tl * WW + k) * D4] = v;                 // imm-offset store
        }
    }
}

extern "C" void kernel_launch(void* const* d_in, const int* in_sizes, int n_in,
                              void* d_out, int out_size, void* d_ws, size_t ws_size,
                              hipStream_t stream) {
    const float* in = (const float*)d_in[0];   // [16, 2048, 256] fp32
    float* out      = (float*)d_out;           // [16, 2048, 7*256] fp32
    (void)in_sizes; (void)n_in; (void)d_ws; (void)ws_size; (void)out_size;

    dim3 grid(BB * (TT / TTILE));   // 2048 blocks
    dim3 block(NTHREADS);           // 8 wave32 waves per block
    window_embed_fwd_kernel<<<grid, block, 0, stream>>>(in, out);
}